// WeightedSumSessEmbedding_69441031242487
// MI455X (gfx1250) — compile-verified
//
#include <hip/hip_runtime.h>

// Ragged per-session mean of gathered embeddings, CDNA5 (gfx1250).
//
// segment_ids sorted -> session s = contiguous range [start,end) via scalar
// binary search. One 64-thread block (2 wave32) per session; wave w owns dims
// [64w, 64w+64), lane owns 4 contiguous dims. Accumulation via four
// independent V_WMMA_F32_16X16X4_F32 with all-ones A: D[m][n] = sum_k B[k][n]
// + C[m][n] -- a free f32 4-item adder tree per WMMA. The sum is invariant to
// the hardware K<->(vgpr,half) mapping of B; only N == lane&15 matters, and
// the per-accumulator dim permutation is undone at the (coalesced b128) store.
// Ids are software-pipelined 2 chunks deep so emb b128 gathers issue from
// resident registers and global_prefetch_b8 (WGP locality) leads by a chunk.

typedef __attribute__((ext_vector_type(2))) float v2f;
typedef __attribute__((ext_vector_type(4))) float v4f;
typedef __attribute__((ext_vector_type(8))) float v8f;

#define EMBED_DIM 128

__device__ __forceinline__ int lower_bound_i32(const int* __restrict__ a, int n, int v) {
    int lo = 0, hi = n;
    while (lo < hi) {
        int mid = (lo + hi) >> 1;        // uniform across block -> SALU/SMEM
        if (a[mid] < v) lo = mid + 1; else hi = mid;
    }
    return lo;
}

__global__ __launch_bounds__(64)
void session_mean_wmma_kernel(const float* __restrict__ emb,
                              const int*   __restrict__ item_ids,
                              const int*   __restrict__ seg,
                              float*       __restrict__ out,
                              int total_items) {
    const int s       = blockIdx.x;
    const int tid     = threadIdx.x;
    const int lane    = tid & 31;
    const int wave    = tid >> 5;
    const int hl      = lane & 15;
    const bool hiHalf = lane >= 16;
    const int dimBase = wave * 64 + hl * 4;   // this lane's 4 dims (acc q -> dim dimBase+q)

    const int start = lower_bound_i32(seg, total_items, s);
    const int end   = lower_bound_i32(seg, total_items, s + 1);

    v8f c0 = {}, c1 = {}, c2 = {}, c3 = {};
    v2f a; a.x = 1.0f; a.y = 1.0f;           // all-ones A => WMMA sums columns

    if (end > start) {
        const int last = end - 1;
        const int off  = hiHalf ? 2 : 0;     // lo half: items i,i+1; hi half: i+2,i+3

        // Pipeline: ids for chunk i (resident) and chunk i+4 (in flight).
        const int iS  = start + off;
        int idA0 = item_ids[min(iS,     last)];
        int idB0 = item_ids[min(iS + 1, last)];
        int idA1 = item_ids[min(iS + 4, last)];
        int idB1 = item_ids[min(iS + 5, last)];

        for (int i = start; i < end; i += 4) {
            const int iA = i + off;
            const int iB = iA + 1;
            const float ma = iA < end ? 1.0f : 0.0f;   // tail masks (no divergence)
            const float mb = iB < end ? 1.0f : 0.0f;

            // Gather this chunk's rows: 16B/lane, 256B contiguous per half-wave.
            v4f va = *(const v4f*)(emb + (size_t)idA0 * EMBED_DIM + dimBase);
            v4f vb = *(const v4f*)(emb + (size_t)idB0 * EMBED_DIM + dimBase);

            // Prefetch next chunk's rows (ids already resident -> no wait here).
            __builtin_prefetch(emb + (size_t)idA1 * EMBED_DIM + dimBase, 0, 3);
            __builtin_prefetch(emb + (size_t)idB1 * EMBED_DIM + dimBase, 0, 3);

            // Ids for chunk i+8 (consumed by next iteration's prefetch).
            const int idA2 = item_ids[min(iA + 8, last)];
            const int idB2 = item_ids[min(iB + 8, last)];

            va.x *= ma; va.y *= ma; va.z *= ma; va.w *= ma;
            vb.x *= mb; vb.y *= mb; vb.z *= mb; vb.w *= mb;

            v2f b0; b0.x = va.x; b0.y = vb.x;
            v2f b1; b1.x = va.y; b1.y = vb.y;
            v2f b2; b2.x = va.z; b2.y = vb.z;
            v2f b3; b3.x = va.w; b3.y = vb.w;

            // Four independent accumulator chains -> WMMAs pipeline freely.
            c0 = __builtin_amdgcn_wmma_f32_16x16x4_f32(false, a, false, b0, (short)0, c0, false, false);
            c1 = __builtin_amdgcn_wmma_f32_16x16x4_f32(false, a, false, b1, (short)0, c1, false, false);
            c2 = __builtin_amdgcn_wmma_f32_16x16x4_f32(false, a, false, b2, (short)0, c2, false, false);
            c3 = __builtin_amdgcn_wmma_f32_16x16x4_f32(false, a, false, b3, (short)0, c3, false, false);

            idA0 = idA1; idB0 = idB1;
            idA1 = idA2; idB1 = idB2;
        }
    }

    const int cnt = end - start;
    const float scale = 1.0f / fmaxf((float)cnt, 1.0f);  // empty session -> 0

    // C/D layout: VGPR0, lanes 0-15 hold row M=0, N=lane. Column hl of acc q
    // is dim dimBase+q -> pack into one coalesced b128 store per lane.
    if (lane < 16) {
        v4f r;
        r.x = c0[0] * scale;
        r.y = c1[0] * scale;
        r.z = c2[0] * scale;
        r.w = c3[0] * scale;
        *(v4f*)(out + (size_t)s * EMBED_DIM + dimBase) = r;
    }
}

extern "C" void kernel_launch(void* const* d_in, const int* in_sizes, int n_in,
                              void* d_out, int out_size, void* d_ws, size_t ws_size,
                              hipStream_t stream) {
    (void)n_in; (void)d_ws; (void)ws_size;
    const float* emb      = (const float*)d_in[0];   // [N_ITEMS, 128] f32
    const int*   item_ids = (const int*)d_in[1];     // [TOTAL_ITEMS] int
    const int*   seg      = (const int*)d_in[2];     // [TOTAL_ITEMS] int, sorted
    // d_in[3] = num_sessions (device scalar) -> derive from out_size instead.
    float* out = (float*)d_out;

    const int total_items  = in_sizes[1];
    const int num_sessions = out_size / EMBED_DIM;

    dim3 grid(num_sessions);
    dim3 block(64);    // 2 wave32, 64 dims per wave
    hipLaunchKernelGGL(session_mean_wmma_kernel, grid, block, 0, stream,
                       emb, item_ids, seg, out, total_items);
}